// SoftRR_48017734369483
// MI455X (gfx1250) — compile-verified
//
#include <hip/hip_runtime.h>
#include <math.h>

// Problem constants (reference: V is 1024 x 8192, num_rounds = 8, tau = 1).
#define NROWS  1024
#define MCOLS  8192
#define STEPS  8192          // num_rounds * NROWS
#define DEPTH  4             // async LDS ring depth (rows prefetched ahead)
#define TPB    1024          // 32 wave32s on one WGP
#define NWAVES (TPB / 32)
#define LOG2E  1.4426950408889634f

// ---------------- CDNA5 async-to-LDS path (guarded) ----------------
typedef int v4i __attribute__((vector_size(16)));

#if __has_builtin(__builtin_amdgcn_global_load_async_to_lds_b128)
#define HAVE_ASYNC 1
#define ASYNC_COPY16(gsrc, ldst)                                           \
  __builtin_amdgcn_global_load_async_to_lds_b128(                          \
      (__attribute__((address_space(1))) v4i*)(void*)(gsrc),               \
      (__attribute__((address_space(3))) v4i*)(void*)(ldst), 0, 0)
#else
#define HAVE_ASYNC 0
#define ASYNC_COPY16(gsrc, ldst)                                           \
  do { *(float4*)(ldst) = *(const float4*)(gsrc); } while (0)
#endif

#if HAVE_ASYNC
#if __has_builtin(__builtin_amdgcn_s_wait_asynccnt)
#define WAIT_ASYNC(n) __builtin_amdgcn_s_wait_asynccnt(n)
#else
#define WAIT_ASYNC(n) asm volatile("s_wait_asynccnt %0" ::"n"(n) : "memory")
#endif
#else
#define WAIT_ASYNC(n) do {} while (0)
#endif

#if __has_builtin(__builtin_amdgcn_exp2f)
#define EXP2F(x) __builtin_amdgcn_exp2f(x)
#else
#define EXP2F(x) exp2f(x)
#endif

#if __has_builtin(__builtin_amdgcn_rcpf)
#define RCPF(x) __builtin_amdgcn_rcpf(x)
#else
#define RCPF(x) (1.0f / (x))
#endif

// ---------------- wave32 shuffle reductions ----------------
__device__ __forceinline__ float wred_sum(float v) {
#pragma unroll
  for (int off = 16; off > 0; off >>= 1)
    v += __shfl_xor(v, off, 32);
  return v;
}
__device__ __forceinline__ float wred_min(float v) {
#pragma unroll
  for (int off = 16; off > 0; off >>= 1)
    v = fminf(v, __shfl_xor(v, off, 32));
  return v;
}

// ---------------- kernel 1: per-row min of V (fully parallel) ----------------
__global__ __launch_bounds__(256) void softrr_rowmin_kernel(
    const float* __restrict__ V, float* __restrict__ rmin) {
  __shared__ float part[8];
  const int row  = blockIdx.x;
  const int tid  = threadIdx.x;
  const int wave = tid >> 5, lane = tid & 31;
  const float* vr = V + (size_t)row * MCOLS;
  float m = 3.402823466e+38f;
#pragma unroll 4
  for (int j = tid; j < MCOLS; j += 256) m = fminf(m, vr[j]);
  m = wred_min(m);
  if (lane == 0) part[wave] = m;
  __syncthreads();
  if (wave == 0) {
    float p = (lane < 8) ? part[lane] : 3.402823466e+38f;
    p = wred_min(p);
    if (lane == 0) rmin[row] = p;
  }
}

// ---------------- kernel 2: the sequential softmax scan ----------------
// One workgroup = 32 wave32s on one WGP. Thread t owns columns [8t, 8t+8).
// Wave w owns the contiguous 1 KB slice of each row it prefetches via async
// copies, so ASYNCcnt (per-wave, in-order) is a sufficient readiness check.
//
// No max-subtraction: x = (v - rowmin + 1)*c with c in (0,1] keeps x in
// [0, ~10] for every step, so exp2(x*log2e) is always in [1, ~2.2e4] and the
// row sum <= ~1.8e8 -- comfortably inside f32 range. Softmax is invariant to
// the shift, so this matches the reference to normal f32 rounding while
// eliminating one of the two per-step workgroup barriers + a reduction tree.
// Carry is held pre-scaled by log2(e): cl = c*log2e, identical recurrence.
__global__ __launch_bounds__(TPB) void softrr_scan_kernel(
    const float* __restrict__ V, const float* __restrict__ rmin,
    float* __restrict__ pi) {
  __shared__ float s_row[DEPTH][MCOLS];   // 128 KB ring of row tiles
  __shared__ float s_rmin[NROWS];         // 4 KB cached row minima
  __shared__ float s_sum[2][NWAVES];      // parity double-buffered partials

  const int tid  = threadIdx.x;
  const int wave = tid >> 5, lane = tid & 31;
  // float4-granularity chunks this lane moves for its wave's 1 KB slice
  const int ch0 = 256 * wave + 4 * lane;  // float index of first 16B chunk
  const int ch1 = ch0 + 128;              // second 16B chunk
  const int colbase = tid * 8;

  for (int i = tid; i < NROWS; i += TPB) s_rmin[i] = rmin[i];
  __syncthreads();

  float cl[8];                            // carry, pre-scaled by log2(e)
#pragma unroll
  for (int k = 0; k < 8; ++k) cl[k] = LOG2E;

  // Prime the ring: rows 0..DEPTH-1 (each wave: 2 async b128 ops per row).
#pragma unroll
  for (int p = 0; p < DEPTH; ++p) {
    const float* g = V + (size_t)p * MCOLS;
    ASYNC_COPY16(g + ch0, &s_row[p][ch0]);
    ASYNC_COPY16(g + ch1, &s_row[p][ch1]);
  }

  for (int r = 0; r < STEPS; ++r) {
    const int row  = r & (NROWS - 1);
    const int slot = r & (DEPTH - 1);
    const int par  = r & 1;

    // Oldest prefetched row must be complete; keep DEPTH-1 rows in flight.
    WAIT_ASYNC((DEPTH - 1) * 2);

    // Kick off the pi-row RMW load early (uniform branch; latency hidden by
    // the reduction below). Round 0 stores fresh values instead.
    float* prow = pi + (size_t)row * MCOLS + colbase;
    const bool rmw = (r >= NROWS);
    float4 o0, o1;
    if (rmw) {
      o0 = *(const float4*)prow;
      o1 = *((const float4*)prow + 1);
    }

    const float bias = 1.0f - s_rmin[row];  // x2 = (v - min + 1) * c * log2e

    // ---- pass 1: x2 = (v + bias) * cl, exponentials, local sum ----
    float4 v0 = *(const float4*)&s_row[slot][colbase];
    float4 v1 = *(const float4*)&s_row[slot][colbase + 4];
    float x2[8];
    x2[0] = (v0.x + bias) * cl[0];
    x2[1] = (v0.y + bias) * cl[1];
    x2[2] = (v0.z + bias) * cl[2];
    x2[3] = (v0.w + bias) * cl[3];
    x2[4] = (v1.x + bias) * cl[4];
    x2[5] = (v1.y + bias) * cl[5];
    x2[6] = (v1.z + bias) * cl[6];
    x2[7] = (v1.w + bias) * cl[7];

    // Slot data consumed -> recycle it: prefetch row r+DEPTH (wave-private
    // slice, so no barrier needed for the reuse).
    {
      const float* g = V + (size_t)((r + DEPTH) & (NROWS - 1)) * MCOLS;
      ASYNC_COPY16(g + ch0, &s_row[slot][ch0]);
      ASYNC_COPY16(g + ch1, &s_row[slot][ch1]);
    }

    float e[8];
    float lsum = 0.0f;
#pragma unroll
    for (int k = 0; k < 8; ++k) {
      e[k] = EXP2F(x2[k]);
      lsum += e[k];
    }

    // ---- single workgroup reduction (sum) ----
    lsum = wred_sum(lsum);
    if (lane == 0) s_sum[par][wave] = lsum;
    __syncthreads();                       // the only barrier per step
    const float gsum = wred_sum(s_sum[par][lane]);
    const float inv  = RCPF(gsum);

    // ---- pass 2: y, carry update, pi accumulate ----
    float y[8];
#pragma unroll
    for (int k = 0; k < 8; ++k) {
      y[k]  = e[k] * inv;
      cl[k] = cl[k] - y[k] * cl[k];        // cl *= (1 - y), one FMA
    }
    if (rmw) {
      o0.x += y[0]; o0.y += y[1]; o0.z += y[2]; o0.w += y[3];
      o1.x += y[4]; o1.y += y[5]; o1.z += y[6]; o1.w += y[7];
    } else {
      o0.x = y[0]; o0.y = y[1]; o0.z = y[2]; o0.w = y[3];
      o1.x = y[4]; o1.y = y[5]; o1.z = y[6]; o1.w = y[7];
    }
    *(float4*)prow       = o0;
    *((float4*)prow + 1) = o1;
  }
}

extern "C" void kernel_launch(void* const* d_in, const int* in_sizes, int n_in,
                              void* d_out, int out_size, void* d_ws,
                              size_t ws_size, hipStream_t stream) {
  (void)in_sizes; (void)n_in; (void)out_size; (void)ws_size;
  const float* V   = (const float*)d_in[0];
  float* rmin      = (float*)d_ws;   // 1024 floats of scratch
  float* pi        = (float*)d_out;  // 1024 x 8192 f32

  softrr_rowmin_kernel<<<NROWS, 256, 0, stream>>>(V, rmin);
  softrr_scan_kernel<<<1, TPB, 0, stream>>>(V, rmin, pi);
}